// TransformLayer_26534307955181
// MI455X (gfx1250) — compile-verified
//
#include <hip/hip_runtime.h>

typedef __attribute__((ext_vector_type(2))) float v2f;
typedef __attribute__((ext_vector_type(8))) float v8f;

#define IMG_W 1024
#define IMG_H 768
#define NBATCH 8
#define HW (IMG_W * IMG_H)
#define EPSV 1e-6f
#define PPT 4          // pixels per thread
#define BLK 256        // threads per block (8 wave32)
#define SPAN (BLK * PPT)

// Fused division-model warp. Each wave32 handles 32 pixels per iteration; the
// 4x4 rigid transform runs on the matrix pipe via V_WMMA_F32_16X16X4_F32.
// Trick: the transform is duplicated into A rows 0-3 AND 8-11, so WMMA#1
// (pixels 0-15) delivers results to lanes 0-15 and WMMA#2 (pixels 16-31)
// delivers results to lanes 16-31 -- every lane gets its own pixel's
// transformed vector with no post-WMMA cross-lane traffic. Only 3 shuffles
// per 32 pixels remain (B-matrix assembly). Memory-bound: 4B in + 16B out.
__global__ __launch_bounds__(BLK) void TransformLayer_kernel(
    const float* __restrict__ inv_depth,   // (B,1,H,W)
    const float* __restrict__ transform,   // (B,4,4)
    const float* __restrict__ calib_i,     // (B,4)
    const float* __restrict__ lam_i_p,     // (B,1)
    const float* __restrict__ calib_j,     // (B,4)
    const float* __restrict__ lam_j_p,     // (B,1)
    float* __restrict__ out_proj,          // (B,3,H,W)
    float* __restrict__ out_mask)          // (B,1,H,W) as 0.0/1.0
{
    const int t    = threadIdx.x;
    const int base = blockIdx.x * SPAN;    // SPAN divides HW -> one batch/block
    const int b    = base / HW;
    const int idx0 = base - b * HW + t;    // pixel index within batch, i = 0
    const int lane = t & 31;
    const bool lo  = lane < 16;

    // ---- per-batch parameters (loaded once per thread) -------------------
    const float fx  = calib_i[b * 4 + 0], fy  = calib_i[b * 4 + 1];
    const float cx  = calib_i[b * 4 + 2], cy  = calib_i[b * 4 + 3];
    const float li  = lam_i_p[b];
    const float fxj = calib_j[b * 4 + 0], fyj = calib_j[b * 4 + 1];
    const float cxj = calib_j[b * 4 + 2], cyj = calib_j[b * 4 + 3];
    const float lj  = lam_j_p[b];
    const float rfx = 1.0f / fx;
    const float rfy = 1.0f / fy;

    // ---- A-matrix: 16x4 f32; transform in rows 0..3 AND 8..11 ------------
    // ISA layout: lanes 0-15 -> M=lane, K={0,1}; lanes 16-31 -> M=lane-16,
    // K={2,3}. V0 = K0/K2, V1 = K1/K3.
    const int m  = lane & 15;
    const int tm = m & 7;                  // rows 8..11 mirror rows 0..3
    const int cb = lo ? 0 : 2;
    v2f A = {0.0f, 0.0f};
    if (tm < 4) {
        A.x = transform[b * 16 + tm * 4 + cb + 0];
        A.y = transform[b * 16 + tm * 4 + cb + 1];
    }

    const v8f c0 = {};

#pragma unroll
    for (int i = 0; i < PPT; ++i) {
        const int idx = idx0 + i * BLK;            // coalesced per iteration
        const int g   = base + t + i * BLK;        // global pixel id
        const int px  = idx & (IMG_W - 1);
        const int py  = idx >> 10;                 // W = 1024

        const float idep = inv_depth[g];

        // ---- undistort (camera i, division model) ------------------------
        const float xn = ((float)px - cx) * rfx;
        const float yn = ((float)py - cy) * rfy;
        const float r2 = xn * xn + yn * yn;
        const float den = 1.0f + li * r2;
        const bool  valid_norm = den > EPSV;
        const float inv_den = 1.0f / (valid_norm ? den : 1.0f);
        const float vx = xn * inv_den;
        const float vy = yn * inv_den;     // point vector (vx, vy, 1, idep)

        // ---- B-matrices: 4x16 f32, columns = pixels ----------------------
        // rows K0/K1 come from lanes 0-15, rows K2/K3 from lanes 16-31.
        const float vx_o = __shfl_xor(vx,   16, 32);  // x of hi-half pixels
        const float vy_o = __shfl_xor(vy,   16, 32);  // y of hi-half pixels
        const float vd_o = __shfl_xor(idep, 16, 32);  // d of lo-half pixels

        v2f B1, B2;                   // B1: pixels 0..15, B2: pixels 16..31
        B1.x = lo ? vx   : 1.0f;      // K0 = x        | K2 = 1
        B1.y = lo ? vy   : vd_o;      // K1 = y        | K3 = inv_depth
        B2.x = lo ? vx_o : 1.0f;
        B2.y = lo ? vy_o : idep;

        // D = A x B: rows 0..3 (lanes 0-15) and rows 8..11 (lanes 16-31)
        // both hold (a,b,c,d); each lane owns its own pixel's result.
        v8f D1 = __builtin_amdgcn_wmma_f32_16x16x4_f32(
            false, A, false, B1, (short)0, c0, false, false);
        v8f D2 = __builtin_amdgcn_wmma_f32_16x16x4_f32(
            false, A, false, B2, (short)0, c0, false, false);

        const float ta = lo ? D1[0] : D2[0];
        const float tb = lo ? D1[1] : D2[1];
        const float tc = lo ? D1[2] : D2[2];
        const float td = lo ? D1[3] : D2[3];

        // ---- perspective divide + distort (camera j) + project -----------
        const bool  mask_src = tc > EPSV;
        const float inv_cz = 1.0f / (mask_src ? tc : 1.0f);
        const float xp  = ta * inv_cz;
        const float yp  = tb * inv_cz;
        const float rho = td * inv_cz;

        const float ru2  = xp * xp + yp * yp;
        const float disc = 1.0f - 4.0f * lj * ru2;
        const bool  valid_un = disc >= 0.0f;
        const float den2 = 2.0f * lj * ru2;
        const bool  use_ratio = valid_un && (fabsf(den2) > EPSV);
        const float sqrt_disc = sqrtf(disc > 0.0f ? disc : 1.0f);
        const float ratio = use_ratio ? (1.0f - sqrt_disc) / den2 : 1.0f;

        const float u = fxj * (xp * ratio) + cxj;
        const float v = fyj * (yp * ratio) + cyj;

        const bool inb = (u >= 0.0f) && (u <= (float)(IMG_W - 1)) &&
                         (v >= 0.0f) && (v <= (float)(IMG_H - 1));
        const bool mask = mask_src && inb && valid_norm && valid_un;

        // ---- stores (coalesced B32 per plane) ----------------------------
        float* op = out_proj + (size_t)b * 3 * HW + idx;
        op[0]       = u;
        op[HW]      = v;
        op[2 * HW]  = rho;
        out_mask[g] = mask ? 1.0f : 0.0f;
    }
}

extern "C" void kernel_launch(void* const* d_in, const int* in_sizes, int n_in,
                              void* d_out, int out_size, void* d_ws, size_t ws_size,
                              hipStream_t stream) {
    const float* inv_depth = (const float*)d_in[0];
    const float* transform = (const float*)d_in[1];
    const float* calib_i   = (const float*)d_in[2];
    const float* lam_i     = (const float*)d_in[3];
    const float* calib_j   = (const float*)d_in[4];
    const float* lam_j     = (const float*)d_in[5];
    // d_in[6] = non_rigid (int scalar) == 0 in this configuration.

    float* out_proj = (float*)d_out;                       // B*3*H*W floats
    float* out_mask = out_proj + (size_t)NBATCH * 3 * HW;  // B*1*H*W floats

    const int total = NBATCH * HW;                         // divisible by SPAN
    hipLaunchKernelGGL(TransformLayer_kernel, dim3(total / SPAN), dim3(BLK), 0,
                       stream, inv_depth, transform, calib_i, lam_i, calib_j,
                       lam_j, out_proj, out_mask);
}